// TrajectoryEncoder_25512105739026
// MI455X (gfx1250) — compile-verified
//
#include <hip/hip_runtime.h>

// ---------------------------------------------------------------------------
// TrajectoryEncoder for MI455X (gfx1250, wave32, WMMA).
// Fixed problem constants from the reference:
static constexpr int D  = 128;   // input dim
static constexpr int H  = 2048;  // hidden dim
static constexpr int DM = 1024;  // d_model
static constexpr int TT = 4;     // number of MLP types (ids T..T+1 are BOT/EOT)
// ---------------------------------------------------------------------------

typedef float v2f __attribute__((ext_vector_type(2)));
typedef float v8f __attribute__((ext_vector_type(8)));

// Full-precision f32 WMMA: D(16x16,f32) = A(16x4,f32) x B(4x16,f32) + C
// 8-arg form per probe: (neg_a, A, neg_b, B, c_mod, C, reuse_a, reuse_b)
__device__ __forceinline__ v8f wmma_f32(v2f a, v2f b, v8f c) {
  return __builtin_amdgcn_wmma_f32_16x16x4_f32(false, a, false, b,
                                               (short)0, c, false, false);
}

// ---------------------------------------------------------------------------
// Workspace layout (as int32):
//   wsI[0..7]   : base[t]  (exclusive prefix of per-type counts; 6 used)
//   wsI[8..15]  : cnt[t]   (per-type counts; 6 used)
//   wsI[16 .. 16+N)       : r[i]     = stable rank of row i (inverse of order)
//   wsI[16+N .. 16+2N)    : order[p] = row index at sorted position p
// ---------------------------------------------------------------------------

#define SORT_THREADS 256

__global__ __launch_bounds__(SORT_THREADS)
void te_sort_kernel(const int* __restrict__ mask, int* __restrict__ wsI, int N) {
  __shared__ int hist[6][SORT_THREADS + 1];
  __shared__ int baseS[6];
  const int tid   = threadIdx.x;
  const int chunk = (N + SORT_THREADS - 1) / SORT_THREADS;
  const int beg   = tid * chunk;
  const int end   = (beg + chunk < N) ? (beg + chunk) : N;

  int h[6] = {0, 0, 0, 0, 0, 0};
  for (int i = beg; i < end; ++i) {
    int t = mask[i];
    h[t]++;
  }
  #pragma unroll
  for (int t = 0; t < 6; ++t) hist[t][tid] = h[t];
  __syncthreads();

  // exclusive scan within each type (6 serial scanners, 256 steps each)
  if (tid < 6) {
    int s = 0;
    for (int i = 0; i < SORT_THREADS; ++i) {
      int v = hist[tid][i];
      hist[tid][i] = s;
      s += v;
    }
    hist[tid][SORT_THREADS] = s;  // total for this type
  }
  __syncthreads();

  if (tid == 0) {
    int b = 0;
    #pragma unroll
    for (int t = 0; t < 6; ++t) {
      baseS[t] = b;
      b += hist[t][SORT_THREADS];
    }
  }
  __syncthreads();

  int* rArr = wsI + 16;
  int* oArr = wsI + 16 + N;

  int cur[6];
  #pragma unroll
  for (int t = 0; t < 6; ++t) cur[t] = baseS[t] + hist[t][tid];

  for (int i = beg; i < end; ++i) {
    int t = mask[i];
    int p = cur[t]++;
    rArr[i] = p;   // rank of row i
    oArr[p] = i;   // row at sorted position p
  }

  if (tid < 6) {
    wsI[tid]     = baseS[tid];
    wsI[8 + tid] = hist[tid][SORT_THREADS];
  }
}

// ---------------------------------------------------------------------------
// Fused MLP: 64 rows of one type per block, 512 threads = 16 wave32s.
//   stage 1: Hc = relu(X[64xD] @ W1[t][:, chunk] + b1)   (chunk of 64 H-cols)
//   stage 2: acc += Hc @ W2[t][chunk, :]                 (64 x DM in VGPRs)
// LDS: XsT (D x 64, padded) + HsT (64 x 64, padded) ~ 51 KB.
// ---------------------------------------------------------------------------

#define HC 64  // H-chunk

__global__ __launch_bounds__(512)
void te_mlp_kernel(const float* __restrict__ x,
                   const float* __restrict__ W1,
                   const float* __restrict__ b1,
                   const float* __restrict__ W2,
                   const float* __restrict__ b2,
                   float* __restrict__ out,
                   const int* __restrict__ wsI, int N) {
  __shared__ float XsT[D][66];       // [k][row]  (pad 66 -> conflict-free)
  __shared__ float HsT[HC][65];      // [kLocal][row]
  __shared__ int gIdx[64];           // gathered source rows
  __shared__ int dIdx[64];           // destination output rows

  const int t       = blockIdx.y;
  const int tile    = blockIdx.x;
  const int base    = wsI[t];
  const int cnt     = wsI[8 + t];
  const int rowBase = tile * 64;
  if (rowBase >= cnt) return;

  const int tid = threadIdx.x;
  const int* rArr = wsI + 16;
  const int* oArr = wsI + 16 + N;

  if (tid < 64) {
    int p     = base + rowBase + tid;
    bool ok   = (rowBase + tid) < cnt;
    int pc    = ok ? p : base;       // clamp to a valid sorted position
    gIdx[tid] = oArr[pc];
    dIdx[tid] = rArr[pc];            // dest(enc row at pos p) == r[p]
  }
  __syncthreads();

  // gather X rows into transposed LDS tile
  for (int idx = tid; idx < 64 * D; idx += 512) {
    int m = idx >> 7;        // row within tile
    int d = idx & (D - 1);   // feature
    XsT[d][m] = x[(size_t)gIdx[m] * D + d];
  }
  __syncthreads();

  const int lane    = tid & 31;
  const int wave    = tid >> 5;            // 0..15
  const int ln      = lane & 15;
  const int koff    = (lane < 16) ? 0 : 2; // K-pair selector (A/B layout)
  const int halfRow = (lane < 16) ? 0 : 8; // D-matrix row half

  const float* W1t = W1 + (size_t)t * D * H;
  const float* W2t = W2 + (size_t)t * H * DM;

  // persistent accumulators: 4 row-tiles x 4 col-tiles = 64 rows x 64 cols/wave
  v8f acc[4][4];
  #pragma unroll
  for (int i = 0; i < 4; ++i)
    #pragma unroll
    for (int j = 0; j < 4; ++j)
      acc[i][j] = (v8f){0.f, 0.f, 0.f, 0.f, 0.f, 0.f, 0.f, 0.f};

  const int s1_rt = wave & 3;   // stage-1 row tile
  const int s1_ct = wave >> 2;  // stage-1 col tile within chunk (0..3)
  const int colW  = wave * 64;  // stage-2 column base for this wave

  for (int c = 0; c < H / HC; ++c) {
    const int hbase = c * HC;

    // ---------------- stage 1: one 16x16 output tile per wave --------------
    {
      const int col0 = hbase + s1_ct * 16;
      v8f a1 = (v8f){0.f, 0.f, 0.f, 0.f, 0.f, 0.f, 0.f, 0.f};
      const float* w1p = W1t + (size_t)koff * H + col0 + ln;
      #pragma unroll 4
      for (int k = 0; k < D; k += 4) {
        v2f A;
        A.x = XsT[k + koff][s1_rt * 16 + ln];
        A.y = XsT[k + koff + 1][s1_rt * 16 + ln];
        v2f B;
        B.x = w1p[(size_t)k * H];         // W1[k+koff  ][col0+ln]
        B.y = w1p[(size_t)(k + 1) * H];   // W1[k+koff+1][col0+ln]
        a1 = wmma_f32(A, B, a1);
      }
      const float bias = b1[t * H + col0 + ln];
      #pragma unroll
      for (int v = 0; v < 8; ++v) {
        float val = a1[v] + bias;
        val = fmaxf(val, 0.0f);  // ReLU
        HsT[s1_ct * 16 + ln][s1_rt * 16 + v + halfRow] = val;
      }
    }
    __syncthreads();

    // ---------------- stage 2: 16 output tiles per wave --------------------
    for (int k = 0; k < HC; k += 4) {
      v2f A[4];
      #pragma unroll
      for (int rt = 0; rt < 4; ++rt) {
        A[rt].x = HsT[k + koff][rt * 16 + ln];
        A[rt].y = HsT[k + koff + 1][rt * 16 + ln];
      }
      const int kg = hbase + k + koff;
      const float* w2p = W2t + (size_t)kg * DM + colW + ln;
      #pragma unroll
      for (int j = 0; j < 4; ++j) {
        v2f B;
        B.x = w2p[j * 16];        // W2[kg  ][colW + j*16 + ln]
        B.y = w2p[j * 16 + DM];   // W2[kg+1][colW + j*16 + ln]
        #pragma unroll
        for (int rt = 0; rt < 4; ++rt)
          acc[rt][j] = wmma_f32(A[rt], B, acc[rt][j]);
      }
    }
    __syncthreads();  // HsT reused next chunk
  }

  // ---------------- epilogue: +b2, scatter rows to out --------------------
  #pragma unroll
  for (int rt = 0; rt < 4; ++rt) {
    #pragma unroll
    for (int j = 0; j < 4; ++j) {
      const int col  = colW + j * 16 + ln;
      const float bias = b2[t * DM + col];
      #pragma unroll
      for (int v = 0; v < 8; ++v) {
        const int m = rt * 16 + v + halfRow;
        if (rowBase + m < cnt) {
          out[(size_t)dIdx[m] * DM + col] = acc[rt][j][v] + bias;
        }
      }
    }
  }
}

// ---------------------------------------------------------------------------
// BOT/EOT rows: out[r[r[j]]] = emb[mask[j] - T]
// ---------------------------------------------------------------------------
__global__ __launch_bounds__(256)
void te_token_kernel(const int* __restrict__ mask,
                     const float* __restrict__ emb,
                     const int* __restrict__ wsI,
                     float* __restrict__ out, int N) {
  const int j = blockIdx.x;
  const int t = mask[j];
  if (t < TT) return;
  const int* rArr = wsI + 16;
  const int dest = rArr[rArr[j]];
  const float* e = emb + (size_t)(t - TT) * DM;
  for (int cidx = threadIdx.x; cidx < DM; cidx += blockDim.x)
    out[(size_t)dest * DM + cidx] = e[cidx];
}

// ---------------------------------------------------------------------------
extern "C" void kernel_launch(void* const* d_in, const int* in_sizes, int n_in,
                              void* d_out, int out_size, void* d_ws, size_t ws_size,
                              hipStream_t stream) {
  const float* x    = (const float*)d_in[0];
  const int*   mask = (const int*)d_in[1];
  const float* W1   = (const float*)d_in[2];
  const float* b1   = (const float*)d_in[3];
  const float* W2   = (const float*)d_in[4];
  const float* b2   = (const float*)d_in[5];
  const float* emb  = (const float*)d_in[6];
  float* out = (float*)d_out;

  const int N = in_sizes[1];
  int* wsI = (int*)d_ws;  // needs (16 + 2N) ints ~ 128 KB for N=16384

  te_sort_kernel<<<1, SORT_THREADS, 0, stream>>>(mask, wsI, N);

  dim3 grid((N + 63) / 64, TT);
  te_mlp_kernel<<<grid, 512, 0, stream>>>(x, W1, b1, W2, b2, out, wsI, N);

  te_token_kernel<<<N, 256, 0, stream>>>(mask, emb, wsI, out, N);
}